// RunoffProducingModel_19086834663458
// MI455X (gfx1250) — compile-verified
//
#include <hip/hip_runtime.h>
#include <hip/hip_bf16.h>

// MI455X (gfx1250) implementation.
// One thread per (b,t) element; 3x3 state in registers (compute-bound kernel,
// ~3.5 KFLOP/elem vs 44 B/elem of HBM traffic). The q/k/val GEMMs (shared
// constant weights across the batch) are batched through V_WMMA_F32_16X16X4_F32:
//   A(16x4)  = [wq^T ; wk^T ; wv^T] stacked in rows (constant per head)
//   B(4x16)  = 16 elements' v[m,0..2] columns (ds_bpermute marshalling)
//   D(16x16) = rows 0..8 are {q,k,val}[m][k] per element-column
// Two A variants (A_lo rows 0..8, A_hi rows 8..15 + row0) make results land in
// the OWNING lane's own VGPRs for both wave halves, so extraction is ~2
// bpermutes per m. Attention/softmax/conv remain per-thread VALU (3x3 blocks
// would be block-diagonal-wasteful under 16x16 WMMA).

typedef float v2f __attribute__((ext_vector_type(2)));
typedef float v8f __attribute__((ext_vector_type(8)));

#define M_LAYERS 3
#define N_DIV 3
#define S_STEPS 3

__device__ __forceinline__ float xl_bperm(int src_lane, float v) {
  return __builtin_bit_cast(
      float, __builtin_amdgcn_ds_bpermute(src_lane << 2,
                                          __builtin_bit_cast(int, v)));
}

__global__ __launch_bounds__(256) void runoff_wmma_kernel(
    const float* __restrict__ rain, const float* __restrict__ evap,
    const float* __restrict__ wm,   const float* __restrict__ we,
    const float* __restrict__ wconv,const float* __restrict__ w11,
    const float* __restrict__ b11,  const float* __restrict__ wq,
    const float* __restrict__ wk,   const float* __restrict__ wv,
    float* __restrict__ out, int BT) {
  const int idx  = blockIdx.x * blockDim.x + threadIdx.x;
  const int cidx = idx < BT ? idx : (BT - 1);   // clamp loads; keep EXEC full
  const int lane = threadIdx.x & 31;
  const bool lo  = lane < 16;
  const int  r15 = lane & 15;

  const float rv   = rain[cidx];
  const float ev   = evap[cidx];
  const float pval = rv * (1.0f / (float)(N_DIV * S_STEPS));

  float wmv[3][3];
#pragma unroll
  for (int m = 0; m < 3; ++m)
#pragma unroll
    for (int n = 0; n < 3; ++n) wmv[m][n] = wm[m * 3 + n];

  float resA[3][3][3];  // per-head results (head0=tanh, head1/2=softmax-9)

#pragma unroll
  for (int h = 0; h < 3; ++h) {
    // ---- uniform (scalar-loaded) per-head weights ----
    float weh[3], wc0[3], wc1[3];
#pragma unroll
    for (int n = 0; n < 3; ++n) {
      weh[n] = we[h * 3 + n];
      wc0[n] = wconv[h * 6 + n];
      wc1[n] = wconv[h * 6 + 3 + n];
    }
    const float s11 = w11[h];
    const float t11 = b11[h];
    float ewn[3];
#pragma unroll
    for (int n = 0; n < 3; ++n) ewn[n] = ev * weh[n];

    // ---- build A_lo / A_hi (16x4 f32 layout: VGPR g -> K = g (lanes<16) or
    //      g+2 (lanes>=16); row = lane&15). Output index o: 0-2=q,3-5=k,6-8=v,
    //      A[o][c] = W_{o/3}[n=c][k=o%3]. A_hi shifts o 0..7 to rows 8..15 and
    //      o=8 to row 0 so upper-half owners read D from their own lanes. ----
    v2f a_lo, a_hi;
#pragma unroll
    for (int g = 0; g < 2; ++g) {
      const int c = lo ? g : (g + 2);
      const int r = r15;
      float vlo = 0.0f;
      if (r < 9 && c < 3) {
        const float* wp = (r < 3) ? wq : (r < 6) ? wk : wv;
        const int k = r - ((r >= 6) ? 6 : (r >= 3) ? 3 : 0);
        vlo = wp[h * 9 + c * 3 + k];
      }
      a_lo[g] = vlo;
      const int o2 = (r >= 8) ? (r - 8) : ((r == 0) ? 8 : -1);
      float vhi = 0.0f;
      if (o2 >= 0 && c < 3) {
        const float* wp = (o2 < 3) ? wq : (o2 < 6) ? wk : wv;
        const int k = o2 - ((o2 >= 6) ? 6 : (o2 >= 3) ? 3 : 0);
        vhi = wp[h * 9 + c * 3 + k];
      }
      a_hi[g] = vhi;
    }

    // ---- state x = sw = 1 * wm ----
    float x[3][3];
#pragma unroll
    for (int m = 0; m < 3; ++m)
#pragma unroll
      for (int n = 0; n < 3; ++n) x[m][n] = wmv[m][n];

#pragma unroll
    for (int s = 0; s < S_STEPS; ++s) {
      // y = concat(x, p) + e*we ; v = 2xN conv then 1x1
      float y[4][3];
#pragma unroll
      for (int m = 0; m < 3; ++m)
#pragma unroll
        for (int n = 0; n < 3; ++n) y[m][n] = x[m][n] + ewn[n];
#pragma unroll
      for (int n = 0; n < 3; ++n) y[3][n] = pval + ewn[n];

      float vv[3][3];
#pragma unroll
      for (int m = 0; m < 3; ++m)
#pragma unroll
        for (int n = 0; n < 3; ++n) {
          float t = fmaf(wc1[n], y[m + 1][n], wc0[n] * y[m][n]);
          vv[m][n] = fmaf(t, s11, t11);
        }

      // ---- q/k/val via WMMA f32 16x16x4, one pair per grid-row m ----
      float q[3][3], kk[3][3], val[3][3];
#pragma unroll
      for (int m = 0; m < 3; ++m) {
        const int srcLo = r15;        // elements owned by lanes 0..15
        const int srcHi = r15 + 16;   // elements owned by lanes 16..31
        // B layout (4x16): VGPR0 = K0 (lanes<16) | K2 (lanes>=16); VGPR1 = K1 | K3(=0)
        const float t0 = xl_bperm(srcLo, vv[m][0]);
        const float t1 = xl_bperm(srcLo, vv[m][1]);
        const float t2 = xl_bperm(srcLo, vv[m][2]);
        v2f bLo;
        bLo[0] = lo ? t0 : t2;
        bLo[1] = lo ? t1 : 0.0f;
        const float u0 = xl_bperm(srcHi, vv[m][0]);
        const float u1 = xl_bperm(srcHi, vv[m][1]);
        const float u2 = xl_bperm(srcHi, vv[m][2]);
        v2f bHi;
        bHi[0] = lo ? u0 : u2;
        bHi[1] = lo ? u1 : 0.0f;

        v8f cz = {};
        v8f dLo = __builtin_amdgcn_wmma_f32_16x16x4_f32(
            false, a_lo, false, bLo, (short)0, cz, false, false);
        v8f dHi = __builtin_amdgcn_wmma_f32_16x16x4_f32(
            false, a_hi, false, bHi, (short)0, cz, false, false);

        // Extraction: outputs 0..7 sit in own lane (dLo for lo half via rows
        // 0..7; dHi for hi half via rows 8..15). Output 8 needs one bpermute.
        float mrg[9];
#pragma unroll
        for (int o = 0; o < 8; ++o) mrg[o] = lo ? dLo[o] : dHi[o];
        const float e8lo = xl_bperm(r15 + 16, dLo[0]);  // row8 -> VGPR0 upper
        const float e8hi = xl_bperm(r15, dHi[0]);       // row0 -> VGPR0 lower
        mrg[8] = lo ? e8lo : e8hi;

        q[m][0] = mrg[0]; q[m][1] = mrg[1]; q[m][2] = mrg[2];
        kk[m][0] = mrg[3]; kk[m][1] = mrg[4]; kk[m][2] = mrg[5];
        val[m][0] = mrg[6]; val[m][1] = mrg[7]; val[m][2] = mrg[8];
      }

      // ---- per-thread attention over the 3 layers ----
      const float scale = 0.5773502691896258f;  // 1/sqrt(3)
      float att[3][3];
#pragma unroll
      for (int i = 0; i < 3; ++i) {
        float l0 = (q[i][0] * kk[0][0] + q[i][1] * kk[0][1] + q[i][2] * kk[0][2]) * scale;
        float l1 = (q[i][0] * kk[1][0] + q[i][1] * kk[1][1] + q[i][2] * kk[1][2]) * scale;
        float l2 = (q[i][0] * kk[2][0] + q[i][1] * kk[2][1] + q[i][2] * kk[2][2]) * scale;
        float mx = fmaxf(l0, fmaxf(l1, l2));
        float e0 = __expf(l0 - mx), e1 = __expf(l1 - mx), e2 = __expf(l2 - mx);
        float inv = 1.0f / (e0 + e1 + e2);
        att[i][0] = e0 * inv; att[i][1] = e1 * inv; att[i][2] = e2 * inv;
      }
      float nx[3][3];
#pragma unroll
      for (int i = 0; i < 3; ++i)
#pragma unroll
        for (int n = 0; n < 3; ++n)
          nx[i][n] = att[i][0] * val[0][n] + att[i][1] * val[1][n] +
                     att[i][2] * val[2][n];
#pragma unroll
      for (int i = 0; i < 3; ++i)
#pragma unroll
        for (int n = 0; n < 3; ++n) x[i][n] = nx[i][n];
    }

    // ---- head activation ----
    if (h == 0) {
#pragma unroll
      for (int m = 0; m < 3; ++m)
#pragma unroll
        for (int n = 0; n < 3; ++n) resA[0][m][n] = tanhf(x[m][n]);
    } else {
      float mx = x[0][0];
#pragma unroll
      for (int m = 0; m < 3; ++m)
#pragma unroll
        for (int n = 0; n < 3; ++n) mx = fmaxf(mx, x[m][n]);
      float sum = 0.0f, ex[3][3];
#pragma unroll
      for (int m = 0; m < 3; ++m)
#pragma unroll
        for (int n = 0; n < 3; ++n) { ex[m][n] = __expf(x[m][n] - mx); sum += ex[m][n]; }
      const float inv = 1.0f / sum;
#pragma unroll
      for (int m = 0; m < 3; ++m)
#pragma unroll
        for (int n = 0; n < 3; ++n) resA[h][m][n] = ex[m][n] * inv;
    }
  }

  // ---- runoff = inf*p - ev_r*e - (swr_new - 1)*wm ; store 9 floats ----
  if (idx < BT) {
    float* op = out + (size_t)idx * 9;
#pragma unroll
    for (int m = 0; m < 3; ++m)
#pragma unroll
      for (int n = 0; n < 3; ++n) {
        float r = resA[1][m][n] * pval - resA[2][m][n] * ev -
                  (resA[0][m][n] - 1.0f) * wmv[m][n];
        op[m * 3 + n] = r;
      }
  }
}

extern "C" void kernel_launch(void* const* d_in, const int* in_sizes, int n_in,
                              void* d_out, int out_size, void* d_ws, size_t ws_size,
                              hipStream_t stream) {
  const float* rain  = (const float*)d_in[0];
  const float* evap  = (const float*)d_in[1];
  const float* wm    = (const float*)d_in[2];
  const float* we    = (const float*)d_in[3];
  const float* wconv = (const float*)d_in[4];
  const float* w11   = (const float*)d_in[5];
  const float* b11   = (const float*)d_in[6];
  const float* wq    = (const float*)d_in[7];
  const float* wk    = (const float*)d_in[8];
  const float* wv    = (const float*)d_in[9];
  float* out = (float*)d_out;

  const int BT = in_sizes[0];          // 512*2048 = 1,048,576 (divisible by 256)
  const int block = 256;
  const int grid = (BT + block - 1) / block;
  runoff_wmma_kernel<<<grid, block, 0, stream>>>(rain, evap, wm, we, wconv, w11,
                                                 b11, wq, wk, wv, out, BT);
}